// DSAFTRMSELoss_7799660609716
// MI455X (gfx1250) — compile-verified
//
#include <hip/hip_runtime.h>
#include <math.h>

// DSAFT RMSE loss, N=16384, single-WGP LDS-resident pipeline for MI455X (gfx1250).
// Phases: (A) e = log(d+eps)-theta, (B) bitonic sort (key,idx) stable via idx tie-break,
// (C) rank = inverse permutation, (D) k = segment-head max-scan, (E) exclusive cumprod,
// (F) prefix-sum of w = e_sorted * d_cdf, (G) residual partials, (H) fp32 WMMA reduction.

#define N_ELEMS 16384
#define THREADS 1024
#define CHUNK   16            // N_ELEMS / THREADS
#define EPSF    1e-32f

typedef __attribute__((ext_vector_type(2))) float v2f;
typedef __attribute__((ext_vector_type(8))) float v8f;

__global__ __launch_bounds__(THREADS)
void dsaft_rmse_kernel(const float* __restrict__ theta,
                       const float* __restrict__ dur,
                       const float* __restrict__ ev,
                       float* __restrict__ out)
{
    __shared__ float          key[N_ELEMS];     // e, then sorted e, then prefix-excl of w
    __shared__ float          cpA[N_ELEMS];     // exclusive cumprod of v
    __shared__ unsigned short idx16[N_ELEMS];   // perm during sort, then k[]
    __shared__ unsigned short rank16[N_ELEMS];  // inverse permutation (rank of original i)
    __shared__ float          fscr[THREADS + 1];
    __shared__ int            iscr[THREADS];

    const int tid  = threadIdx.x;
    const int base = tid * CHUNK;

    // ---- Phase A: keys + identity indices (coalesced) ----
    for (int j = tid; j < N_ELEMS; j += THREADS) {
        key[j]   = logf(dur[j] + EPSF) - theta[j];
        idx16[j] = (unsigned short)j;
    }

    // ---- Phase B: bitonic sort ascending on (key, idx) ----
    for (unsigned size = 2; size <= N_ELEMS; size <<= 1) {
        for (unsigned stride = size >> 1; stride > 0; stride >>= 1) {
            __syncthreads();
            #pragma unroll
            for (int q = 0; q < (N_ELEMS / 2) / THREADS; ++q) {
                unsigned p = (unsigned)tid + (unsigned)q * THREADS;
                unsigned i = (p << 1) - (p & (stride - 1));
                unsigned j = i + stride;
                bool up = ((i & size) == 0);
                float ki = key[i], kj = key[j];
                unsigned short ii = idx16[i], jj = idx16[j];
                bool gt = (ki > kj) || ((ki == kj) && (ii > jj));
                if (gt == up) {
                    key[i] = kj;  key[j] = ki;
                    idx16[i] = jj; idx16[j] = ii;
                }
            }
        }
    }
    __syncthreads();

    // ---- Phase C: rank = inverse permutation ----
    for (int p = tid; p < N_ELEMS; p += THREADS)
        rank16[idx16[p]] = (unsigned short)p;
    __syncthreads();

    // ---- Phase D: k[p] = segment-head index (searchsorted-left), via max-scan ----
    int localK[CHUNK];
    {
        int runmax = -1;
        #pragma unroll
        for (int t = 0; t < CHUNK; ++t) {
            int j = base + t;
            float kj = key[j];
            bool head = (j == 0) || (key[j - 1] != kj);
            if (head) runmax = j;
            localK[t] = runmax;
        }
        iscr[tid] = runmax;
    }
    __syncthreads();
    if (tid == 0) {
        int carry = -1;
        for (int t = 0; t < THREADS; ++t) {
            int tmp = iscr[t]; iscr[t] = carry; carry = carry > tmp ? carry : tmp;
        }
    }
    __syncthreads();
    {
        int carry = iscr[tid];
        #pragma unroll
        for (int t = 0; t < CHUNK; ++t) {
            int kv = (localK[t] >= 0) ? localK[t] : carry;
            idx16[base + t] = (unsigned short)kv;   // idx16 now holds k[]
        }
    }
    __syncthreads();

    // ---- Phase E: cpA = exclusive cumprod of v[j] = |1 - ev_s[j]/(n-k[j])| ----
    float lv[CHUNK];
    {
        float prod = 1.0f;
        #pragma unroll
        for (int t = 0; t < CHUNK; ++t) {
            int j  = base + t;
            int kj = (int)idx16[j];
            float evj = ev[(int)rank16[j]];
            float vv  = fabsf(1.0f - evj / (float)(N_ELEMS - kj));
            lv[t] = prod;
            prod *= vv;
        }
        fscr[tid] = prod;
    }
    __syncthreads();
    if (tid == 0) {
        float carry = 1.0f;
        for (int t = 0; t < THREADS; ++t) {
            float tmp = fscr[t]; fscr[t] = carry; carry *= tmp;
        }
    }
    __syncthreads();
    {
        float cb = fscr[tid];
        #pragma unroll
        for (int t = 0; t < CHUNK; ++t) cpA[base + t] = cb * lv[t];
    }
    __syncthreads();

    // ---- Phase F: prefix-excl sum of w[j] = e_s[j]*d_cdf[j]; overwrite key[] ----
    float lw[CHUNK];
    {
        float sum = 0.0f;
        #pragma unroll
        for (int t = 0; t < CHUNK; ++t) {
            int j = base + t;
            float Fj = fabsf(cpA[(int)idx16[j]] - 1.0f);
            float dcdf;
            if (j == N_ELEMS - 1) dcdf = 1.0f - Fj;
            else                  dcdf = fabsf(cpA[(int)idx16[j + 1]] - 1.0f) - Fj;
            lw[t] = sum;
            sum  += key[j] * dcdf;
        }
        fscr[tid] = sum;
    }
    __syncthreads();
    if (tid == 0) {
        float carry = 0.0f;
        for (int t = 0; t < THREADS; ++t) {
            float tmp = fscr[t]; fscr[t] = carry; carry += tmp;
        }
        fscr[THREADS] = carry;  // total_w
    }
    __syncthreads();
    float wbase   = fscr[tid];
    float total_w = fscr[THREADS];
    {
        #pragma unroll
        for (int t = 0; t < CHUNK; ++t) key[base + t] = wbase + lw[t];
    }
    __syncthreads();

    // ---- Phase G: residual partial sums ----
    {
        float acc = 0.0f;
        #pragma unroll
        for (int t = 0; t < CHUNK; ++t) {
            int j  = base + t;
            int kj = (int)idx16[j];
            float surv = cpA[kj];
            float rsv  = total_w - key[kj];       // suffix sum rs[k[j]]
            float cE   = rsv / surv;
            int r = (int)rank16[j];
            float evj   = ev[r];
            float resid = evj * (logf(dur[r] + EPSF) - cE - theta[r]) + cE;
            acc += resid * resid;
        }
        fscr[tid] = acc;
    }
    __syncthreads();

    // ---- Phase H: exact fp32 WMMA reduction of 1024 partials (wave 0 only) ----
    // D(16x16) = A(16x4 partials) x B(4x16 ones) + C ; ones are exact in f32, so
    // the accumulation is pure fp32 adds. Sum all 256 accumulator entries /16
    // (layout-agnostic: each A row-sum appears once per column).
    if (tid < 32) {                       // wave-uniform: EXEC all ones in wave 0
        v8f c = {};
        v2f b; b[0] = 1.0f; b[1] = 1.0f;
        #pragma unroll
        for (int it = 0; it < 16; ++it) {
            v2f a;
            a[0] = fscr[it * 64 + tid];
            a[1] = fscr[it * 64 + 32 + tid];
            c = __builtin_amdgcn_wmma_f32_16x16x4_f32(
                    /*neg_a=*/false, a, /*neg_b=*/false, b,
                    /*c_mod=*/(short)0, c, /*reuse_a=*/false, /*reuse_b=*/false);
        }
        float s = c[0] + c[1] + c[2] + c[3] + c[4] + c[5] + c[6] + c[7];
        s += __shfl_xor(s, 16, 32);
        s += __shfl_xor(s, 8, 32);
        s += __shfl_xor(s, 4, 32);
        s += __shfl_xor(s, 2, 32);
        s += __shfl_xor(s, 1, 32);
        if (tid == 0) out[0] = sqrtf((s * (1.0f / 16.0f)) / (float)N_ELEMS);
    }
}

extern "C" void kernel_launch(void* const* d_in, const int* in_sizes, int n_in,
                              void* d_out, int out_size, void* d_ws, size_t ws_size,
                              hipStream_t stream) {
    const float* theta = (const float*)d_in[0];  // log_h (N,1) -> flat N
    const float* dur   = (const float*)d_in[1];  // durations (N,)
    const float* ev    = (const float*)d_in[2];  // events (N,)
    float* out = (float*)d_out;
    dsaft_rmse_kernel<<<1, THREADS, 0, stream>>>(theta, dur, ev, out);
}